// SentenceEncoder_37125697307047
// MI455X (gfx1250) — compile-verified
//
#include <hip/hip_runtime.h>

#define LPAD 8192
#define HSZ  25
#define ESZ  50
#define G4   100   // 4*H

typedef __attribute__((ext_vector_type(2))) float v2f;
typedef __attribute__((ext_vector_type(8))) float v8f;

__device__ __forceinline__ float readlane_f(float v, int lane) {
  return __builtin_bit_cast(float, __builtin_amdgcn_readlane(__builtin_bit_cast(int, v), lane));
}

__device__ __forceinline__ float fast_sigmoid(float x) {
  return 1.0f / (1.0f + __expf(-x));
}

// ---------------------------------------------------------------------------
// Phase 1: zs[dir][t][j] = Emb[tok][:] @ Wi[:, j] + b[j]  via V_WMMA_F32_16X16X4_F32
// grid = (512 M-tiles, 7 N-tiles, 2 dirs), block = 32 (one wave per 16x16 tile)
//
// Branch-free operand gather: indices are clamped (always-valid loads) and
// out-of-range values zeroed with selects (v_cndmask), so no EXEC-mask
// branching appears around the loads. All 52 fragment loads are issued before
// the WMMA chain; two independent accumulator chains hide the WMMA->WMMA RAW
// hazard NOPs.
// ---------------------------------------------------------------------------
__global__ void __launch_bounds__(32)
zs_wmma_kernel(const int* __restrict__ sentence, const int* __restrict__ slen_p,
               const float* __restrict__ E_f, const float* __restrict__ Wi_f, const float* __restrict__ b_f,
               const float* __restrict__ E_b, const float* __restrict__ Wi_b, const float* __restrict__ b_b,
               float* __restrict__ zs)   // [2][LPAD][G4]
{
  const int lane    = threadIdx.x;      // 0..31
  const int mt      = blockIdx.x;       // 0..511  (rows of 16)
  const int nt      = blockIdx.y;       // 0..6    (cols of 16, covers 100 padded to 112)
  const int dir     = blockIdx.z;       // 0 fwd, 1 bwd
  const int length  = *slen_p;

  const float* Emb = dir ? E_b  : E_f;
  const float* Wi  = dir ? Wi_b : Wi_f;
  const float* bia = dir ? b_b  : b_f;

  const int halfSel = lane >> 4;        // 0: lanes 0-15, 1: lanes 16-31
  const int lid     = lane & 15;

  // Token row for this lane's A slice (both wave halves cover the same 16 rows,
  // but different K pairs per the 32-bit A-matrix layout).
  const int t = mt * 16 + lid;
  int tok;
  if (dir == 0) {
    tok = sentence[t];
  } else {
    const int idx = (t < length) ? (length - 1 - t) : (LPAD - 1 - (t - length));
    tok = sentence[idx];
  }
  const float* erow = Emb + (long)tok * ESZ;

  const int  n0    = nt * 16;
  const int  ncol  = n0 + lid;                    // logical B/N column of this lane
  const bool colok = (ncol < G4);
  const int  ncolc = colok ? ncol : (G4 - 1);     // clamped (always-valid) column

  // ---- Gather all fragments (branch-free, loads issued together) ----
  v2f afrag[13], bfrag[13];
#pragma unroll
  for (int kk = 0; kk < 13; ++kk) {
    const int  kbase = kk * 4 + halfSel * 2;      // this lane covers kbase, kbase+1
    const bool kok   = (kbase < ESZ);             // kbase even => kbase+1 < ESZ too
    const int  k0    = kok ? kbase : 0;           // clamped indices, always valid
    const int  k1    = kok ? (kbase + 1) : 0;

    const float a0 = erow[k0];
    const float a1 = erow[k1];
    const float w0 = Wi[k0 * G4 + ncolc];
    const float w1 = Wi[k1 * G4 + ncolc];

    const bool ok = kok && colok;
    afrag[kk].x = kok ? a0 : 0.0f;
    afrag[kk].y = kok ? a1 : 0.0f;
    bfrag[kk].x = ok ? w0 : 0.0f;
    bfrag[kk].y = ok ? w1 : 0.0f;
  }

  // ---- 13 WMMAs: two independent accumulator chains (hazard hiding) ----
  v8f acc0 = {};
  v8f acc1 = {};
#pragma unroll
  for (int kk = 0; kk < 13; kk += 2) {
    acc0 = __builtin_amdgcn_wmma_f32_16x16x4_f32(
        false, afrag[kk], false, bfrag[kk], (short)0, acc0, false, false);
    if (kk + 1 < 13)
      acc1 = __builtin_amdgcn_wmma_f32_16x16x4_f32(
          false, afrag[kk + 1], false, bfrag[kk + 1], (short)0, acc1, false, false);
  }

  // ---- Bias add + store. C/D layout: VGPR r -> row (mt*16 + r + 8*halfSel) ----
  if (colok) {
    const float bias = bia[ncol];
    float* zrow = zs + (((long)dir * LPAD) + mt * 16 + 8 * halfSel) * G4 + ncol;
#pragma unroll
    for (int r = 0; r < 8; ++r) {
      zrow[(long)r * G4] = (acc0[r] + acc1[r]) + bias;
    }
  }
}

// ---------------------------------------------------------------------------
// Phase 2: sequential LSTM scan. One wave32 per direction (2 blocks).
// Lane l (< 25) owns cell l: accumulators for gates i/f/g/o at columns
// l, 25+l, 50+l, 75+l. Wh held in ~100 VGPRs per lane; h broadcast via
// v_readlane (single wave => no barriers needed).
// ---------------------------------------------------------------------------
__global__ void __launch_bounds__(32)
lstm_scan_kernel(const int* __restrict__ slen_p,
                 const float* __restrict__ Wh_f, const float* __restrict__ Wh_b,
                 const float* __restrict__ zs,   // [2][LPAD][G4]
                 float* __restrict__ out)        // [LPAD][2*HSZ]
{
  const int lane   = threadIdx.x;
  const int dir    = blockIdx.x;
  const int length = *slen_p;

  const float* Wh = dir ? Wh_b : Wh_f;
  const float* z  = zs + (long)dir * LPAD * G4;
  float* o        = out + dir * HSZ;

  const bool active = (lane < HSZ);
  const int  l      = active ? lane : 0;

  // Preload this lane's four Wh columns: wh{g}[k] = Wh[k][25*g + l]
  float wh0[HSZ], wh1[HSZ], wh2[HSZ], wh3[HSZ];
#pragma unroll
  for (int k = 0; k < HSZ; ++k) {
    wh0[k] = Wh[k * G4 + l];
    wh1[k] = Wh[k * G4 + HSZ + l];
    wh2[k] = Wh[k * G4 + 2 * HSZ + l];
    wh3[k] = Wh[k * G4 + 3 * HSZ + l];
  }

  float c = 0.0f, h = 0.0f;

  // Software pipeline: z-row loads issued one step ahead.
  float zi = z[l], zf = z[HSZ + l], zg = z[2 * HSZ + l], zo = z[3 * HSZ + l];

  for (int t = 0; t < LPAD; ++t) {
    // Issue next step's loads early (clamped at the end).
    const int tn = (t + 1 < LPAD) ? (t + 1) : t;
    const float* zn = z + (long)tn * G4;
    float nzi = zn[l];
    float nzf = zn[HSZ + l];
    float nzg = zn[2 * HSZ + l];
    float nzo = zn[3 * HSZ + l];
    // Warm the near cache a few rows ahead (-> global_prefetch_b8).
    const int tp = (t + 8 < LPAD) ? (t + 8) : (LPAD - 1);
    __builtin_prefetch(z + (long)tp * G4, 0, 3);

    // z_t + h @ Wh : four independent 25-FMA chains, h broadcast by readlane.
    float gi = zi, gf = zf, gg = zg, go = zo;
#pragma unroll
    for (int k = 0; k < HSZ; ++k) {
      const float hk = readlane_f(h, k);
      gi = fmaf(hk, wh0[k], gi);
      gf = fmaf(hk, wh1[k], gf);
      gg = fmaf(hk, wh2[k], gg);
      go = fmaf(hk, wh3[k], go);
    }

    const float ig = fast_sigmoid(gi);
    const float fg = fast_sigmoid(gf);
    const float gt = tanhf(gg);
    const float og = fast_sigmoid(go);

    const float nc = fg * c + ig * gt;
    const float nh = og * tanhf(nc);

    // Reference emits the UNMASKED nh every step...
    if (active) o[(long)t * (2 * HSZ) + lane] = nh;

    // ...but the carry freezes past sentence_length.
    const bool keep = (t < length);
    c = keep ? nc : c;
    h = keep ? nh : h;

    zi = nzi; zf = nzf; zg = nzg; zo = nzo;
  }
}

// ---------------------------------------------------------------------------
extern "C" void kernel_launch(void* const* d_in, const int* in_sizes, int n_in,
                              void* d_out, int out_size, void* d_ws, size_t ws_size,
                              hipStream_t stream) {
  (void)in_sizes; (void)n_in; (void)out_size; (void)ws_size;

  const int*   sentence = (const int*)d_in[0];
  const int*   slen     = (const int*)d_in[1];
  const float* E_f      = (const float*)d_in[2];
  const float* Wi_f     = (const float*)d_in[3];
  const float* Wh_f     = (const float*)d_in[4];
  const float* b_f      = (const float*)d_in[5];
  const float* E_b      = (const float*)d_in[6];
  const float* Wi_b     = (const float*)d_in[7];
  const float* Wh_b     = (const float*)d_in[8];
  const float* b_b      = (const float*)d_in[9];

  float* zs = (float*)d_ws;              // 2 * 8192 * 100 f32 = 6.55 MB scratch

  // Phase 1: input projections for both directions via f32 WMMA.
  dim3 g1(LPAD / 16, (G4 + 15) / 16, 2); // 512 x 7 x 2 waves
  zs_wmma_kernel<<<g1, 32, 0, stream>>>(sentence, slen,
                                        E_f, Wi_f, b_f,
                                        E_b, Wi_b, b_b, zs);

  // Phase 2: both directional scans in parallel (one wave each).
  lstm_scan_kernel<<<2, 32, 0, stream>>>(slen, Wh_f, Wh_b, zs, (float*)d_out);
}